// OdyssNet_17944373362812
// MI455X (gfx1250) — compile-verified
//
#include <hip/hip_runtime.h>
#include <math.h>

#define NN     4096
#define NIN    32
#define NOUT   32
#define BATCH  32
#define TSTEPS 64
#define KSPLIT 8
#define KCHUNK (NN / KSPLIT)   // 512
#define EPSV   1e-6f

typedef __attribute__((ext_vector_type(2))) float v2f;
typedef __attribute__((ext_vector_type(8))) float v8f;

// ---------------------------------------------------------------------------
// init: state = state0 ; s_inj = state0 with x[:,0,:] injected at input_pos
// ---------------------------------------------------------------------------
__global__ __launch_bounds__(256) void k_init(
    const float* __restrict__ state0, const float* __restrict__ x,
    const float* __restrict__ in_scale, const int* __restrict__ ipos,
    float* __restrict__ state, float* __restrict__ s_inj)
{
  int b = blockIdx.x;
  for (int n = threadIdx.x; n < NN; n += blockDim.x) {
    float v = state0[b * NN + n];
    state[b * NN + n] = v;
    s_inj[b * NN + n] = v;
  }
  __syncthreads();
  if (threadIdx.x < NIN) {
    int j = threadIdx.x;
    int p = ipos[j];
    s_inj[b * NN + p] += x[(b * TSTEPS + 0) * NIN + j] * in_scale[j];
  }
}

// ---------------------------------------------------------------------------
// Split-K fp32 WMMA GEMM: pre_part[ks][b][n] = sum_{k in slice} s_inj[b][k]*W[n][k]
// One wave per (32-wide N supertile, K-slice). Each wave holds:
//   - 2 A fragments (M rows 0-15 and 16-31 of the batch) shared across N tiles
//   - 2 B fragments (two adjacent 16-wide N tiles of W)
//   - 4 f32 16x16 accumulators
// => 4 global_load_b64 per 4 v_wmma per K-step of 4 (1.0 VMEM / WMMA).
//   A 16x4 f32 fragment: lanes 0-15 rows M, VGPR{0,1}=K{k,k+1}; lanes 16-31 K{k+2,k+3}
//   B 4x16 f32 fragment: lanes 0-15 cols N, VGPR{0,1}=K{k,k+1}; lanes 16-31 K{k+2,k+3}
//   C/D 16x16 f32: VGPR r: lanes 0-15 -> M=r, lanes 16-31 -> M=r+8; lane&15 = N
// ---------------------------------------------------------------------------
__global__ __launch_bounds__(256) void k_gemm(
    const float* __restrict__ s_inj, const float* __restrict__ W,
    float* __restrict__ pre_part)
{
  int wid  = (int)((blockIdx.x * blockDim.x + threadIdx.x) >> 5);
  int lane = threadIdx.x & 31;
  int ntile2 = wid & (NN / 32 - 1);   // 0..127 (32-column supertile)
  int kslice = wid >> 7;              // 0..KSPLIT-1

  int n0      = ntile2 * 32;
  int lane_lo = lane & 15;
  int khalf   = (lane >> 4) * 2;      // 0 for lanes 0-15, 2 for lanes 16-31
  int kbase   = kslice * KCHUNK;

  const float* aptr0 = s_inj + (size_t)lane_lo * NN + khalf;          // M rows 0..15
  const float* aptr1 = s_inj + (size_t)(lane_lo + 16) * NN + khalf;   // M rows 16..31
  const float* bptr0 = W + (size_t)(n0 + lane_lo) * NN + khalf;       // N tile 0
  const float* bptr1 = W + (size_t)(n0 + 16 + lane_lo) * NN + khalf;  // N tile 1

  v8f c00 = {};   // M 0-15 , N tile 0
  v8f c10 = {};   // M 16-31, N tile 0
  v8f c01 = {};   // M 0-15 , N tile 1
  v8f c11 = {};   // M 16-31, N tile 1

#pragma unroll 4
  for (int k = kbase; k < kbase + KCHUNK; k += 4) {
    v2f a0 = *(const v2f*)(aptr0 + k);   // 8B-aligned global_load_b64
    v2f a1 = *(const v2f*)(aptr1 + k);
    v2f b0 = *(const v2f*)(bptr0 + k);
    v2f b1 = *(const v2f*)(bptr1 + k);
    c00 = __builtin_amdgcn_wmma_f32_16x16x4_f32(false, a0, false, b0,
                                                (short)0, c00, false, false);
    c10 = __builtin_amdgcn_wmma_f32_16x16x4_f32(false, a1, false, b0,
                                                (short)0, c10, false, false);
    c01 = __builtin_amdgcn_wmma_f32_16x16x4_f32(false, a0, false, b1,
                                                (short)0, c01, false, false);
    c11 = __builtin_amdgcn_wmma_f32_16x16x4_f32(false, a1, false, b1,
                                                (short)0, c11, false, false);
  }

  float* op = pre_part + (size_t)kslice * BATCH * NN;
  int col = n0 + lane_lo;
  int mh  = (lane >> 4) * 8;
#pragma unroll
  for (int r = 0; r < 8; ++r) {
    op[(size_t)(r + mh) * NN + col]           = c00[r];
    op[(size_t)(16 + r + mh) * NN + col]      = c10[r];
    op[(size_t)(r + mh) * NN + col + 16]      = c01[r];
    op[(size_t)(16 + r + mh) * NN + col + 16] = c11[r];
  }
}

// ---------------------------------------------------------------------------
// Fused: sum K-partials + bias -> tanh -> + memory feedback -> RMSNorm ->
// new state, next injected state, output slice. One block per batch row.
// ---------------------------------------------------------------------------
__global__ __launch_bounds__(256) void k_norm(
    const float* __restrict__ pre_part, const float* __restrict__ bias,
    const float* __restrict__ mfb, const float* __restrict__ mgate,
    const float* __restrict__ norm_w, const float* __restrict__ x,
    const float* __restrict__ in_scale, const float* __restrict__ out_scale,
    const int* __restrict__ ipos, const int* __restrict__ opos,
    float* __restrict__ state, float* __restrict__ s_inj,
    float* __restrict__ out, int t)
{
  __shared__ float red[256];
  int b = blockIdx.x;

  float ss = 0.f;
  for (int n = threadIdx.x; n < NN; n += 256) {
    float pre = bias[n];
#pragma unroll
    for (int ks = 0; ks < KSPLIT; ++ks)
      pre += pre_part[(size_t)ks * BATCH * NN + (size_t)b * NN + n];
    float h   = tanhf(pre);
    float mem = tanhf(state[b * NN + n] * mfb[n]) * mgate[n];
    float tv  = h + mem;
    s_inj[b * NN + n] = tv;          // stash pre-norm value (s_inj is free now)
    ss += tv * tv;
  }
  red[threadIdx.x] = ss;
  __syncthreads();
  for (int off = 128; off > 0; off >>= 1) {
    if (threadIdx.x < off) red[threadIdx.x] += red[threadIdx.x + off];
    __syncthreads();
  }
  float scale = rsqrtf(red[0] * (1.0f / (float)NN) + EPSV);

  for (int n = threadIdx.x; n < NN; n += 256) {
    float ns = s_inj[b * NN + n] * scale * norm_w[n];
    state[b * NN + n] = ns;
    s_inj[b * NN + n] = ns;          // becomes next step's GEMM input
  }
  __syncthreads();

  if (threadIdx.x < NOUT) {
    int j = threadIdx.x;
    out[(b * TSTEPS + t) * NOUT + j] = state[b * NN + opos[j]] * out_scale[j];
    if (t + 1 < TSTEPS) {
      int p = ipos[j];
      s_inj[b * NN + p] += x[(b * TSTEPS + (t + 1)) * NIN + j] * in_scale[j];
    }
  }
}

// ---------------------------------------------------------------------------
extern "C" void kernel_launch(void* const* d_in, const int* in_sizes, int n_in,
                              void* d_out, int out_size, void* d_ws, size_t ws_size,
                              hipStream_t stream)
{
  const float* x      = (const float*)d_in[0];
  const float* W      = (const float*)d_in[1];
  const float* bias   = (const float*)d_in[2];
  const float* mfb    = (const float*)d_in[3];
  const float* mgate  = (const float*)d_in[4];
  const float* iscale = (const float*)d_in[5];
  const float* oscale = (const float*)d_in[6];
  const float* norm_w = (const float*)d_in[7];
  const float* state0 = (const float*)d_in[8];
  const int*   ipos   = (const int*)d_in[9];
  const int*   opos   = (const int*)d_in[10];
  float* out = (float*)d_out;

  float* ws       = (float*)d_ws;
  float* s_inj    = ws;                           // B*N floats
  float* state    = ws + (size_t)BATCH * NN;      // B*N floats
  float* pre_part = ws + (size_t)2 * BATCH * NN;  // KSPLIT*B*N floats

  k_init<<<BATCH, 256, 0, stream>>>(state0, x, iscale, ipos, state, s_inj);

  const int gemm_waves  = KSPLIT * (NN / 32);           // 1024 waves
  const int gemm_blocks = gemm_waves * 32 / 256;        // 128 blocks of 8 waves

  for (int t = 0; t < TSTEPS; ++t) {
    k_gemm<<<gemm_blocks, 256, 0, stream>>>(s_inj, W, pre_part);
    k_norm<<<BATCH, 256, 0, stream>>>(pre_part, bias, mfb, mgate, norm_w, x,
                                      iscale, oscale, ipos, opos,
                                      state, s_inj, out, t);
  }
}